// Random_MMF_43576738185286
// MI455X (gfx1250) — compile-verified
//
#include <hip/hip_runtime.h>
#include <hip/hip_bf16.h>

typedef __attribute__((ext_vector_type(2))) float v2f;
typedef __attribute__((ext_vector_type(8))) float v8f;

static constexpr int    NDIM = 4096;
static constexpr int    LVLS = 128;
static constexpr size_t NN   = (size_t)NDIM * NDIM;       // 16777216
static constexpr int    NACT = 3968;                       // N - L*DROP

// d_out layout (floats)
static constexpr size_t OFF_AREC   = 0;
static constexpr size_t OFF_D      = NN;                   // 16777216
static constexpr size_t OFF_WAVE   = 2 * NN;               // 33554432
static constexpr size_t OFF_CORE   = OFF_WAVE + (size_t)LVLS * NDIM;
static constexpr size_t OFF_MOTHER = OFF_CORE + (size_t)NACT * NDIM;
static constexpr size_t OFF_FATHER = OFF_MOTHER + (size_t)LVLS * NDIM;

// ---------------- init kernels ----------------
__global__ void k_init_identity(float* __restrict__ R) {
  size_t idx = (size_t)blockIdx.x * blockDim.x + threadIdx.x;
  int i = (int)(idx >> 12), j = (int)(idx & 4095);
  R[idx] = (i == j) ? 1.0f : 0.0f;
}

__global__ void k_zero_flags(float* __restrict__ f) {
  int i = blockIdx.x * blockDim.x + threadIdx.x;
  if (i < NDIM) f[i] = 0.0f;
}

__global__ void k_set_flags(float* __restrict__ f, const int* __restrict__ act) {
  int i = blockIdx.x * blockDim.x + threadIdx.x;
  if (i < NACT) f[act[i]] = 1.0f;
}

// ---------------- scan kernels (one level) ----------------
// A[sel,:] = O @ A[sel,:]   and  right[sel,:] = O @ right[sel,:]  (blockIdx.y picks matrix)
__global__ __launch_bounds__(256) void k_level_rows(float* __restrict__ Aw,
                                                    float* __restrict__ R,
                                                    const float* __restrict__ Os,
                                                    const int* __restrict__ sel, int l) {
  __shared__ float sO[256];
  __shared__ int   sSel[16];
  int tid = threadIdx.x;
  sO[tid] = Os[l * 256 + tid];
  if (tid < 16) sSel[tid] = sel[l * 16 + tid];
  __syncthreads();
  float* M = (blockIdx.y == 0) ? Aw : R;
  int c = blockIdx.x * 256 + tid;
  float v[16];
#pragma unroll
  for (int j = 0; j < 16; ++j) v[j] = M[(size_t)sSel[j] * NDIM + c];
#pragma unroll
  for (int i = 0; i < 16; ++i) {
    float s = 0.0f;
#pragma unroll
    for (int j = 0; j < 16; ++j) s = fmaf(sO[i * 16 + j], v[j], s);
    M[(size_t)sSel[i] * NDIM + c] = s;
  }
}

// A[:,sel] = A[:,sel] @ O^T  -> per row r: A[r,sel_i] = sum_j O[i,j]*A[r,sel_j]
__global__ __launch_bounds__(256) void k_level_cols(float* __restrict__ Aw,
                                                    const float* __restrict__ Os,
                                                    const int* __restrict__ sel, int l) {
  __shared__ float sO[256];
  __shared__ int   sSel[16];
  int tid = threadIdx.x;
  sO[tid] = Os[l * 256 + tid];
  if (tid < 16) sSel[tid] = sel[l * 16 + tid];
  __syncthreads();
  int r = blockIdx.x * 256 + tid;
  float v[16];
#pragma unroll
  for (int j = 0; j < 16; ++j) v[j] = Aw[(size_t)r * NDIM + sSel[j]];
#pragma unroll
  for (int i = 0; i < 16; ++i) {
    float s = 0.0f;
#pragma unroll
    for (int j = 0; j < 16; ++j) s = fmaf(sO[i * 16 + j], v[j], s);
    Aw[(size_t)r * NDIM + sSel[i]] = s;
  }
}

__global__ void k_wavelet(const float* __restrict__ Aw, const int* __restrict__ drp,
                          float* __restrict__ Wout, int l) {
  int c = blockIdx.x * blockDim.x + threadIdx.x;
  int row = drp[l];
  Wout[(size_t)l * NDIM + c] = Aw[(size_t)row * NDIM + c];
}

// ---------------- epilogue helpers ----------------
__global__ void k_maskD(const float* __restrict__ Aw, const float* __restrict__ flags,
                        float* __restrict__ D) {
  size_t idx = (size_t)blockIdx.x * blockDim.x + threadIdx.x;
  int i = (int)(idx >> 12), j = (int)(idx & 4095);
  float m = (i == j) ? 1.0f : flags[i] * flags[j];
  D[idx] = Aw[idx] * m;
}

__global__ void k_gather_rows(const float* __restrict__ src, const int* __restrict__ idx,
                              float* __restrict__ dst) {
  int c = blockIdx.x * 256 + threadIdx.x;
  int r = blockIdx.y;
  dst[(size_t)r * NDIM + c] = src[(size_t)idx[r] * NDIM + c];
}

// ---------------- fp32 WMMA GEMM:  C = op(A) * B  (op = I or transpose) ----------------
// 128x128 C-block / workgroup, 8 waves, each wave 64x32 (4x2 tiles of 16x16).
// K-panel of 32 in LDS; A stored [m][k], B stored transposed [n][k] so both
// fragments are one ds_load_b64 per lane (ISA f32 16x16x4 layout).
__global__ __launch_bounds__(256) void k_gemm_wmma(const float* __restrict__ Am,
                                                   const float* __restrict__ Bm,
                                                   float* __restrict__ Cm, int transA) {
  constexpr int LD = 36; // floats; keeps 16B alignment, spreads banks
  __shared__ float As[128 * LD];
  __shared__ float Bs[128 * LD];
  const int tid   = threadIdx.x;
  const int lane  = tid & 31;
  const int wave  = tid >> 5;
  const int waveM = wave >> 2;  // 0..1
  const int waveN = wave & 3;   // 0..3
  const int h     = lane >> 4;  // lane half
  const int l16   = lane & 15;
  const int mBase = blockIdx.y * 128;
  const int nBase = blockIdx.x * 128;

  v8f acc[4][2];
#pragma unroll
  for (int a = 0; a < 4; ++a)
#pragma unroll
    for (int b = 0; b < 2; ++b) acc[a][b] = (v8f)0.0f;

  for (int kb = 0; kb < NDIM; kb += 32) {
    if (!transA) {
#pragma unroll
      for (int i = 0; i < 4; ++i) {
        int q = tid + i * 256;
        int row = q >> 3;
        int kq = (q & 7) * 4;
        const float4 v = *(const float4*)(Am + (size_t)(mBase + row) * NDIM + kb + kq);
        *(float4*)(&As[row * LD + kq]) = v;
      }
    } else { // As[m][k] = Am[k][mBase+m]  (transpose during fill)
#pragma unroll
      for (int i = 0; i < 4; ++i) {
        int q = tid + i * 256;
        int k = q >> 5;
        int mq = (q & 31) * 4;
        const float4 v = *(const float4*)(Am + (size_t)(kb + k) * NDIM + mBase + mq);
        As[(mq + 0) * LD + k] = v.x;
        As[(mq + 1) * LD + k] = v.y;
        As[(mq + 2) * LD + k] = v.z;
        As[(mq + 3) * LD + k] = v.w;
      }
    }
    // B panel, stored transposed: Bs[n][k] = Bm[k][nBase+n]
#pragma unroll
    for (int i = 0; i < 4; ++i) {
      int q = tid + i * 256;
      int k = q >> 5;
      int nq = (q & 31) * 4;
      const float4 v = *(const float4*)(Bm + (size_t)(kb + k) * NDIM + nBase + nq);
      Bs[(nq + 0) * LD + k] = v.x;
      Bs[(nq + 1) * LD + k] = v.y;
      Bs[(nq + 2) * LD + k] = v.z;
      Bs[(nq + 3) * LD + k] = v.w;
    }
    __syncthreads();

#pragma unroll
    for (int kk = 0; kk < 32; kk += 4) {
      v2f afrag[4], bfrag[2];
#pragma unroll
      for (int mt = 0; mt < 4; ++mt)
        afrag[mt] = *(const v2f*)(&As[(waveM * 64 + mt * 16 + l16) * LD + kk + 2 * h]);
#pragma unroll
      for (int nt = 0; nt < 2; ++nt)
        bfrag[nt] = *(const v2f*)(&Bs[(waveN * 32 + nt * 16 + l16) * LD + kk + 2 * h]);
#pragma unroll
      for (int mt = 0; mt < 4; ++mt)
#pragma unroll
        for (int nt = 0; nt < 2; ++nt)
          acc[mt][nt] = __builtin_amdgcn_wmma_f32_16x16x4_f32(
              false, afrag[mt], false, bfrag[nt], (short)0, acc[mt][nt], false, false);
    }
    __syncthreads();
  }

  // C/D layout: VGPR v -> row (v + 8*laneHalf), col = lane&15
#pragma unroll
  for (int mt = 0; mt < 4; ++mt)
#pragma unroll
    for (int nt = 0; nt < 2; ++nt) {
      int col = nBase + waveN * 32 + nt * 16 + l16;
#pragma unroll
      for (int v = 0; v < 8; ++v) {
        int row = mBase + waveM * 64 + mt * 16 + v + 8 * h;
        Cm[(size_t)row * NDIM + col] = acc[mt][nt][v];
      }
    }
}

// ---------------- driver ----------------
extern "C" void kernel_launch(void* const* d_in, const int* in_sizes, int n_in,
                              void* d_out, int out_size, void* d_ws, size_t ws_size,
                              hipStream_t stream) {
  const float* A    = (const float*)d_in[0];
  const float* Os   = (const float*)d_in[1];
  const int*   sel  = (const int*)d_in[2];
  const int*   drp  = (const int*)d_in[3];
  const int*   act  = (const int*)d_in[4];
  const int*   inac = (const int*)d_in[5];
  float* out = (float*)d_out;
  float* ws  = (float*)d_ws;

  float* A_work = ws;             // NN
  float* rightM = ws + NN;        // NN
  float* M1     = ws + 2 * NN;    // NN
  float* flags  = ws + 3 * NN;    // NDIM

  // init: A_work = A ; right = I ; active flags
  hipMemcpyAsync(A_work, A, NN * sizeof(float), hipMemcpyDeviceToDevice, stream);
  k_init_identity<<<NN / 256, 256, 0, stream>>>(rightM);
  k_zero_flags<<<NDIM / 256, 256, 0, stream>>>(flags);
  k_set_flags<<<(NACT + 255) / 256, 256, 0, stream>>>(flags, act);

  // sequential multiresolution scan
  for (int l = 0; l < LVLS; ++l) {
    k_level_rows<<<dim3(NDIM / 256, 2), 256, 0, stream>>>(A_work, rightM, Os, sel, l);
    k_level_cols<<<NDIM / 256, 256, 0, stream>>>(A_work, Os, sel, l);
    k_wavelet<<<NDIM / 256, 256, 0, stream>>>(A_work, drp, out + OFF_WAVE, l);
  }

  // D = A_L * mask (into output slot; also GEMM input)
  k_maskD<<<NN / 256, 256, 0, stream>>>(A_work, flags, out + OFF_D);

  // row gathers
  k_gather_rows<<<dim3(NDIM / 256, NACT), 256, 0, stream>>>(A_work, act, out + OFF_CORE);
  k_gather_rows<<<dim3(NDIM / 256, LVLS), 256, 0, stream>>>(rightM, inac, out + OFF_MOTHER);
  k_gather_rows<<<dim3(NDIM / 256, NACT), 256, 0, stream>>>(rightM, act, out + OFF_FATHER);

  // A_rec = right^T @ (D @ right)  — two 4096^3 fp32 WMMA GEMMs
  k_gemm_wmma<<<dim3(32, 32), 256, 0, stream>>>(out + OFF_D, rightM, M1, 0);
  k_gemm_wmma<<<dim3(32, 32), 256, 0, stream>>>(rightM, M1, out + OFF_AREC, 1);
}